// LoRA_layer_34394098106459
// MI455X (gfx1250) — compile-verified
//
#include <hip/hip_runtime.h>
#include <stdint.h>

// ---- problem constants (match reference) ----
#define IN_DIM   2048
#define OUT_DIM  2048
#define S_DIM    2048
#define B_DIM    16
#define RANKV    8
#define SCALE    2.0f

// ---- tiling ----
#define WAVES          8
#define ROWS_PER_WAVE  16
#define ROWS_PER_BLOCK (WAVES * ROWS_PER_WAVE)   // 128 rows / block
#define CK             64                        // K-chunk staged per wave
#define CKP            68                        // padded LDS row stride (floats):
                                                 // 4*row % 64 spreads lanes over all banks

typedef float v2f __attribute__((ext_vector_type(2)));
typedef float v4f __attribute__((ext_vector_type(4)));
typedef float v8f __attribute__((ext_vector_type(8)));

// low 32 bits of a generic pointer into LDS == byte offset in wave's LDS window
__device__ __forceinline__ uint32_t lds_off(const void* p) {
  return (uint32_t)(uintptr_t)p;
}

// CDNA5 async DMA: global -> LDS, tracked by ASYNCcnt
__device__ __forceinline__ void async_copy_b128(uint32_t lds_byte_off, const void* gaddr) {
  asm volatile("global_load_async_to_lds_b128 %0, %1, off"
               :: "v"(lds_byte_off), "v"(gaddr)
               : "memory");
}
// same, non-temporal (x is read exactly once)
__device__ __forceinline__ void async_copy_b128_nt(uint32_t lds_byte_off, const void* gaddr) {
  asm volatile("global_load_async_to_lds_b128 %0, %1, off th:TH_LOAD_NT"
               :: "v"(lds_byte_off), "v"(gaddr)
               : "memory");
}
__device__ __forceinline__ void wait_async_le8() {
  asm volatile("s_wait_asynccnt 8" ::: "memory");
}
__device__ __forceinline__ void wait_async_0() {
  asm volatile("s_wait_asynccnt 0" ::: "memory");
}

__global__ __launch_bounds__(256) void lora_fused_kernel(
    const float* __restrict__ x,
    const int*   __restrict__ idx,
    const float* __restrict__ lora_a,
    const float* __restrict__ lora_b,
    float*       __restrict__ out)
{
  __shared__ float sA[IN_DIM * RANKV];                    // 64 KB: A[pool] (k-major, [k][r])
  __shared__ float sX[2 * WAVES * ROWS_PER_WAVE * CKP];   // ~68 KB: double-buffered x staging

  const int tid  = threadIdx.x;
  const int lane = tid & 31;
  const int wave = tid >> 5;
  const int half = lane >> 4;     // 0: lanes 0-15, 1: lanes 16-31
  const int lm   = lane & 15;

  const int blk  = blockIdx.x;               // 256 blocks
  const int bb   = blk >> 4;                 // batch element (16 blocks each)
  const int srow = (blk & 15) * ROWS_PER_BLOCK;
  const int pool = idx[bb];

  const float* xb = x      + ((size_t)bb * S_DIM + srow) * IN_DIM;
  const float* Ap = lora_a + (size_t)pool * IN_DIM * RANKV;
  const float* Bp = lora_b + (size_t)pool * RANKV * OUT_DIM;
  float*       ob = out    + ((size_t)bb * S_DIM + srow) * OUT_DIM;

  // ---- stage 0: DMA A[pool] (64KB) into LDS, 64 floats / thread, coalesced b128 ----
  #pragma unroll
  for (int t = 0; t < 16; ++t) {
    const int i = (t * 256 + tid) * 4;
    async_copy_b128(lds_off(&sA[i]), Ap + i);
  }

  const int wrow = wave * ROWS_PER_WAVE;
  float* sX0 = &sX[(size_t)wave * ROWS_PER_WAVE * CKP];
  float* sX1 = &sX[(size_t)(WAVES + wave) * ROWS_PER_WAVE * CKP];

  // per-wave x-chunk stager: 16 rows x 64 floats, 8 b128 DMAs per lane (NT: read-once)
  auto stage_x = [&](float* dst, int kc) {
    #pragma unroll
    for (int t = 0; t < 8; ++t) {
      const int row = 2 * t + half;
      const int k   = lm * 4;
      async_copy_b128_nt(lds_off(&dst[row * CKP + k]),
                         xb + (size_t)(wrow + row) * IN_DIM + kc + k);
    }
  };

  // prefetch first chunk, then drain ALL asyncs (covers sA too) and sync waves for sA
  stage_x(sX0, 0);
  wait_async_0();
  __syncthreads();

  // ---- stage 1: XA[16x16] = X[16x2048] @ A[2048x8]  (cols 8..15 exactly zero) ----
  v8f acc0 = {};
  v8f acc1 = {};
  const float bmask = (lm < 8) ? 1.0f : 0.0f;   // rank lanes 8..15 feed zeros
  const int   rsel  = lm & 7;

  int p = 0;
  for (int kc = 0; kc < IN_DIM; kc += CK, p ^= 1) {
    float* cur = p ? sX1 : sX0;
    float* nxt = p ? sX0 : sX1;
    if (kc + CK < IN_DIM) {
      stage_x(nxt, kc + CK);    // overlap DMA of next chunk with WMMA on current
      wait_async_le8();         // previous chunk (first 8 in flight) complete
    } else {
      wait_async_0();
    }

    #pragma unroll
    for (int k0 = 0; k0 < CK; k0 += 8) {
      // A-operand (16x4 f32): lane = row, {half,vgpr} = k   -> conflict-free ds_load_b64
      v2f a0 = *(const v2f*)&cur[lm * CKP + k0 + 0 + 2 * half];
      v2f a1 = *(const v2f*)&cur[lm * CKP + k0 + 4 + 2 * half];
      // B-operand (4x16 f32): lane = rank, {half,vgpr} = k  -> broadcast-pair LDS reads
      const int kb0 = kc + k0 + 0 + 2 * half;
      const int kb1 = kc + k0 + 4 + 2 * half;
      v2f b0, b1;
      b0.x = sA[(kb0 + 0) * RANKV + rsel] * bmask;
      b0.y = sA[(kb0 + 1) * RANKV + rsel] * bmask;
      b1.x = sA[(kb1 + 0) * RANKV + rsel] * bmask;
      b1.y = sA[(kb1 + 1) * RANKV + rsel] * bmask;
      acc0 = __builtin_amdgcn_wmma_f32_16x16x4_f32(false, a0, false, b0,
                                                   (short)0, acc0, false, false);
      acc1 = __builtin_amdgcn_wmma_f32_16x16x4_f32(false, a1, false, b1,
                                                   (short)0, acc1, false, false);
    }
  }

  // ---- D-layout -> K-striped relayout of XA through private LDS scratch; fold 2.0x ----
  float* sXA = sX0;   // wave-private, reuse staging buffer 0 (stride 16 floats)
  #pragma unroll
  for (int v = 0; v < 8; ++v) {
    const int row = v + 8 * half;          // D layout: M = vgpr + 8*half, N(rank) = lm
    sXA[row * 16 + lm] = (acc0[v] + acc1[v]) * SCALE;
  }
  // XA as stage-2 *B-operand*: lane = row, {half,vgpr} = rank
  v2f xa0 = *(const v2f*)&sXA[lm * 16 + 0 + 2 * half];   // ranks 0..3
  v2f xa1 = *(const v2f*)&sXA[lm * 16 + 4 + 2 * half];   // ranks 4..7

  // ---- stage 2 (transposed): OUT^T-tile = B-tile x XA^T ----
  // A-operand = B_lora tile (lane = outcol-in-tile, {half,vgpr} = rank)  [L2-hot]
  // D layout:  D[outcol = v + 8*half][row = lane%16]
  //   -> each lane holds 8 CONSECUTIVE output columns of one row: 2x b128 NT stores
  for (int nt = 0; nt < OUT_DIM / 16; ++nt) {
    const int n = nt * 16 + lm;
    v2f b0, b1;
    b0.x = Bp[(size_t)(0 + 2 * half + 0) * OUT_DIM + n];
    b0.y = Bp[(size_t)(0 + 2 * half + 1) * OUT_DIM + n];
    b1.x = Bp[(size_t)(4 + 2 * half + 0) * OUT_DIM + n];
    b1.y = Bp[(size_t)(4 + 2 * half + 1) * OUT_DIM + n];
    v8f o = {};
    o = __builtin_amdgcn_wmma_f32_16x16x4_f32(false, b0, false, xa0,
                                              (short)0, o, false, false);
    o = __builtin_amdgcn_wmma_f32_16x16x4_f32(false, b1, false, xa1,
                                              (short)0, o, false, false);
    // lane writes out[wrow + lm][nt*16 + 8*half + 0..7]
    float* dst = ob + (size_t)(wrow + lm) * OUT_DIM + nt * 16 + 8 * half;
    v4f s0 = { o[0], o[1], o[2], o[3] };
    v4f s1 = { o[4], o[5], o[6], o[7] };
    __builtin_nontemporal_store(s0, (v4f*)dst);
    __builtin_nontemporal_store(s1, (v4f*)(dst + 4));
  }
}

extern "C" void kernel_launch(void* const* d_in, const int* in_sizes, int n_in,
                              void* d_out, int out_size, void* d_ws, size_t ws_size,
                              hipStream_t stream) {
  (void)in_sizes; (void)n_in; (void)out_size; (void)d_ws; (void)ws_size;
  const float* x      = (const float*)d_in[0];
  const int*   idx    = (const int*)d_in[1];
  const float* lora_a = (const float*)d_in[2];
  const float* lora_b = (const float*)d_in[3];
  float*       out    = (float*)d_out;

  const int blocks = (B_DIM * S_DIM) / ROWS_PER_BLOCK;   // 256
  hipLaunchKernelGGL(lora_fused_kernel, dim3(blocks), dim3(256), 0, stream,
                     x, idx, lora_a, lora_b, out);
}